// RadialDCLayerSingleCoil_72181220376636
// MI455X (gfx1250) — compile-verified
//
#include <hip/hip_runtime.h>
#include <math.h>

typedef float v2f __attribute__((ext_vector_type(2)));
typedef float v8f __attribute__((ext_vector_type(8)));

#define IM 256
#define NSP 288
#define NSAMP 640
#define M_TOT (NSP * NSAMP)          // 184320
#define ORTHO (1.0f / 256.0f)        // 1/sqrt(256*256)
#define KSTEPS 64                    // 256 / 4 (K-steps over v for forward GEMM)
#define NTILES_U 16                  // 256 / 16 column tiles
#define ADJ_CHUNKS 16
#define M_PER_CHUNK (M_TOT / ADJ_CHUNKS)   // 11520
#define ADJ_KSTEPS (M_PER_CHUNK / 4)       // 2880

// ---- CDNA5 f32 WMMA: D(16x16) = A(16x4) * B(4x16) + C ----
__device__ __forceinline__ v8f wmma_f32(v2f a, v2f b, v8f c) {
    return __builtin_amdgcn_wmma_f32_16x16x4_f32(
        /*neg_a=*/false, a, /*neg_b=*/false, b,
        /*c_mod=*/(short)0, c, /*reuse_a=*/false, /*reuse_b=*/false);
}

// ------------------------------------------------------------------
// Kernel 1: repack image into WMMA B-fragment layout.
// B[v][u] = img[u][v] (plane p: 0=re, 1=im), fragment for k-step kv,
// u-tile tu, lane l holds {B[4kv+2h+0][u], B[4kv+2h+1][u]}, u = 16*tu + (l&15).
// Flat index: ((p*16 + tu)*64 + kv)*32 + lane  (float2 each)
// ------------------------------------------------------------------
__global__ void pack_img(const float* __restrict__ x, float2* __restrict__ bpack) {
    int t = blockIdx.x * blockDim.x + threadIdx.x;   // 65536 threads
    int lane = t & 31;
    int kv   = (t >> 5) & 63;
    int tu   = (t >> 11) & 15;
    int p    = t >> 15;
    int h    = lane >> 4;
    int iu   = 16 * tu + (lane & 15);
    int iv0  = 4 * kv + 2 * h;
    float v0 = x[((iu << 8) + iv0) * 2 + p];
    float v1 = x[((iu << 8) + iv0 + 1) * 2 + p];
    bpack[t] = make_float2(v0, v1);
}

// ------------------------------------------------------------------
// Kernel 2: forward NUFFT (16 samples per wave) + dcf + sigmoid blend.
// A[m] = ORTHO * sum_u e^{-i kx u} * (sum_v img[u,v] e^{-i ky v})
// k_dc[m] = lam*A[m]*dcf[m] + (1-lam)*y[m%640, m/640]
// ------------------------------------------------------------------
__global__ void __launch_bounds__(32) fwd_nufft(
    const float* __restrict__ ktraj, const float* __restrict__ y,
    const float* __restrict__ lambda_p, const float2* __restrict__ bpack,
    float* __restrict__ kdc) {

    __shared__ float4 eyS[KSTEPS * 32];              // 32 KB: Ey fragments

    int m0   = blockIdx.x * 16;
    int lane = threadIdx.x;
    int row  = lane & 15;
    int h    = lane >> 4;
    const float* kx = ktraj;
    const float* ky = ktraj + M_TOT;

    // Build Ey A-fragments: Ey = cos(th) - i sin(th), th = ky_m * (iv-128)
    float kyv = ky[m0 + row];
    for (int kv = 0; kv < KSTEPS; ++kv) {
        int iv = 4 * kv + 2 * h;
        float s0, c0, s1, c1;
        __sincosf(kyv * (float)(iv - 128), &s0, &c0);
        __sincosf(kyv * (float)(iv - 127), &s1, &c1);
        eyS[kv * 32 + lane] = make_float4(c0, c1, s0, s1);
    }
    __syncthreads();

    float kxv[8];
#pragma unroll
    for (int r = 0; r < 8; ++r) kxv[r] = kx[m0 + r + 8 * h];

    float sumR[8], sumI[8];
#pragma unroll
    for (int r = 0; r < 8; ++r) { sumR[r] = 0.0f; sumI[r] = 0.0f; }

    for (int tu = 0; tu < NTILES_U; ++tu) {
        v8f accTr  = {};   // c@gr + s@gi
        v8f accTiA = {};   // c@gi
        v8f accTiB = {};   // s@gr   (Ti = A - B)
        const float2* bre = bpack + (size_t)tu * (KSTEPS * 32);
        const float2* bim = bpack + (size_t)(16 + tu) * (KSTEPS * 32);
        for (int kv = 0; kv < KSTEPS; ++kv) {
            float4 ey = eyS[kv * 32 + lane];
            v2f cfr = { ey.x, ey.y };
            v2f sfr = { ey.z, ey.w };
            float2 g0 = bre[kv * 32 + lane];
            float2 g1 = bim[kv * 32 + lane];
            v2f bgr = { g0.x, g0.y };
            v2f bgi = { g1.x, g1.y };
            // T2 = (c - i s)(gr + i gi) = (c*gr + s*gi) + i(c*gi - s*gr)
            accTr  = wmma_f32(cfr, bgr, accTr);
            accTr  = wmma_f32(sfr, bgi, accTr);
            accTiA = wmma_f32(cfr, bgi, accTiA);
            accTiB = wmma_f32(sfr, bgr, accTiB);
        }
        // Multiply T2[m,u] by Ex = cos - i sin, accumulate over u
#pragma unroll
        for (int r = 0; r < 8; ++r) {
            float Tr = accTr[r];
            float Ti = accTiA[r] - accTiB[r];
            float u  = (float)(16 * tu + row - 128);
            float sn, cs;
            __sincosf(kxv[r] * u, &sn, &cs);
            sumR[r] += cs * Tr + sn * Ti;
            sumI[r] += cs * Ti - sn * Tr;
        }
    }

    // Reduce over the N dimension (16 lanes per half-wave)
#pragma unroll
    for (int r = 0; r < 8; ++r) {
        for (int ofs = 1; ofs < 16; ofs <<= 1) {
            sumR[r] += __shfl_xor(sumR[r], ofs, 16);
            sumI[r] += __shfl_xor(sumI[r], ofs, 16);
        }
    }

    float lam = 1.0f / (1.0f + __expf(-lambda_p[0]));
#pragma unroll
    for (int r = 0; r < 8; ++r) {
        if (row == r) {
            int m = m0 + r + 8 * h;
            float Ar = sumR[r] * ORTHO;
            float Ai = sumI[r] * ORTHO;
            float kxm = kx[m], kym = ky[m];
            float dcf = sqrtf(kxm * kxm + kym * kym);
            Ar *= dcf; Ai *= dcf;
            int s = m % NSAMP, p = m / NSAMP;
            float yr = y[(s * NSP + p) * 2 + 0];
            float yi = y[(s * NSP + p) * 2 + 1];
            kdc[2 * m + 0] = lam * Ar + (1.0f - lam) * yr;
            kdc[2 * m + 1] = lam * Ai + (1.0f - lam) * yi;
        }
    }
}

// ------------------------------------------------------------------
// Kernel 3: adjoint NUFFT. One wave per (16x16 output tile, K-chunk).
// x[u,v] += sum_m e^{+i kx u} * k_dc[m] * e^{+i ky v}
// ------------------------------------------------------------------
__global__ void __launch_bounds__(32) adj_nufft(
    const float* __restrict__ ktraj, const float* __restrict__ kdc,
    float* __restrict__ partials) {

    int tile  = blockIdx.x;   // 0..255
    int chunk = blockIdx.y;   // 0..ADJ_CHUNKS-1
    int lane  = threadIdx.x;
    int row   = lane & 15;
    int h     = lane >> 4;
    int iu0   = (tile >> 4) * 16;
    int iv0   = (tile & 15) * 16;
    const float* kx = ktraj;
    const float* ky = ktraj + M_TOT;

    float u = (float)(iu0 + row - 128);   // A-fragment row coordinate
    float v = (float)(iv0 + row - 128);   // B-fragment column coordinate
    int mbase = chunk * M_PER_CHUNK;

    v8f accRa = {}, accRb = {}, accIa = {}, accIb = {};
    for (int k = 0; k < ADJ_KSTEPS; ++k) {
        int mb = mbase + 4 * k + 2 * h;
        float2 kx2 = *(const float2*)(kx + mb);
        float2 ky2 = *(const float2*)(ky + mb);
        float4 kd  = *(const float4*)(kdc + 2 * mb);   // (re0,im0,re1,im1)

        float sa0, ca0, sa1, ca1;
        __sincosf(kx2.x * u, &sa0, &ca0);
        __sincosf(kx2.y * u, &sa1, &ca1);
        v2f ar = { ca0, ca1 };
        v2f ai = { sa0, sa1 };

        float sb0, cb0, sb1, cb1;
        __sincosf(ky2.x * v, &sb0, &cb0);
        __sincosf(ky2.y * v, &sb1, &cb1);
        // B = k_dc * e^{+i phi}
        v2f br = { kd.x * cb0 - kd.y * sb0, kd.z * cb1 - kd.w * sb1 };
        v2f bi = { kd.x * sb0 + kd.y * cb0, kd.z * sb1 + kd.w * cb1 };

        // X = (Ar + i Ai)(Br + i Bi): Xr = Ar@Br - Ai@Bi ; Xi = Ar@Bi + Ai@Br
        accRa = wmma_f32(ar, br, accRa);
        accRb = wmma_f32(ai, bi, accRb);
        accIa = wmma_f32(ar, bi, accIa);
        accIb = wmma_f32(ai, br, accIb);
    }

    float* out = partials + (size_t)(tile * ADJ_CHUNKS + chunk) * 512;
#pragma unroll
    for (int r = 0; r < 8; ++r) {
        float Xr = accRa[r] - accRb[r];
        float Xi = accIa[r] + accIb[r];
        int e = (r + 8 * h) * 16 + row;   // (iu_local, iv_local)
        out[2 * e + 0] = Xr;
        out[2 * e + 1] = Xi;
    }
}

// ------------------------------------------------------------------
// Kernel 4: sum K-chunk partials, apply ortho scale, write output image.
// ------------------------------------------------------------------
__global__ void finalize(const float* __restrict__ partials, float* __restrict__ out) {
    int t  = blockIdx.x * blockDim.x + threadIdx.x;   // 65536
    int iu = t >> 8, iv = t & 255;
    int tile = (iu >> 4) * 16 + (iv >> 4);
    int e    = (iu & 15) * 16 + (iv & 15);
    float sr = 0.0f, si = 0.0f;
    for (int c = 0; c < ADJ_CHUNKS; ++c) {
        const float* p = partials + (size_t)(tile * ADJ_CHUNKS + c) * 512 + 2 * e;
        sr += p[0];
        si += p[1];
    }
    out[2 * t + 0] = sr * ORTHO;
    out[2 * t + 1] = si * ORTHO;
}

extern "C" void kernel_launch(void* const* d_in, const int* in_sizes, int n_in,
                              void* d_out, int out_size, void* d_ws, size_t ws_size,
                              hipStream_t stream) {
    (void)in_sizes; (void)n_in; (void)out_size; (void)ws_size;
    const float* x     = (const float*)d_in[0];   // (1,1,256,256,2)
    const float* y     = (const float*)d_in[1];   // (640,288,2)
    const float* lam   = (const float*)d_in[2];   // (1,)
    const float* ktraj = (const float*)d_in[3];   // (1,2,184320)

    char*   ws       = (char*)d_ws;
    float2* bpack    = (float2*)ws;                              // 512 KB
    float*  kdc      = (float*)(ws + 524288);                    // 1.44 MB
    float*  partials = (float*)(ws + 524288 + 1474560);          // 8 MB

    pack_img<<<256, 256, 0, stream>>>(x, bpack);
    fwd_nufft<<<M_TOT / 16, 32, 0, stream>>>(ktraj, y, lam, bpack, kdc);
    adj_nufft<<<dim3(256, ADJ_CHUNKS), 32, 0, stream>>>(ktraj, kdc, partials);
    finalize<<<256, 256, 0, stream>>>(partials, (float*)d_out);
}